// SWD_28449863369545
// MI455X (gfx1250) — compile-verified
//
#include <hip/hip_runtime.h>
#include <hip/hip_bf16.h>

// Problem constants (B=4, H=8, S=1024, D=64)
#define N_MAT   32          // B*H
#define S_LEN   1024
#define D_DIM   64
#define OUT_ELEMS 33554432u // N_MAT * S * S
#define SORT_BLOCK 512      // 16 waves: 2 CAS per thread per stage
#define BLOCK   256

// ---- CDNA5 async global->LDS path (probe-confirmed in rounds 1-3) ----
#if defined(__gfx1250__) && __has_builtin(__builtin_amdgcn_global_load_async_to_lds_b32) && __has_builtin(__builtin_amdgcn_s_wait_asynccnt)
#define USE_ASYNC_LDS 1
typedef __attribute__((address_space(1))) int* gbl_i32p;
typedef __attribute__((address_space(3))) int* lds_i32p;
#else
#define USE_ASYNC_LDS 0
#endif

// Wave-local LDS sync: DS ops from one wave complete in order, so for
// stages whose producer/consumer lanes stay inside one wave32 we only
// need a compiler fence + s_wait_dscnt, not a full s_barrier.
__device__ __forceinline__ void wave_lds_sync() {
#if defined(__gfx1250__) && __has_builtin(__builtin_amdgcn_s_wait_dscnt) && __has_builtin(__builtin_amdgcn_wave_barrier)
    __builtin_amdgcn_wave_barrier();
    __builtin_amdgcn_s_wait_dscnt(0);
    __builtin_amdgcn_wave_barrier();
#else
    __syncthreads();
#endif
}

// ------------------------------------------------------------------
// Kernel A: one block per (n, d) column pair.
//   - gather q/k columns (stride 256B) into LDS via
//     GLOBAL_LOAD_ASYNC_TO_LDS_B32 (ASYNCcnt), then s_wait_asynccnt
//   - bitonic sort both (value, index) arrays; full barrier only when
//     the stage's pair distance (or the previous one) crosses a wave
//   - w = exp(-(qs-ks)^2)/64, scatter atomicAdd into out[n, qi, ki]
// LDS: 16KB of the 320KB WGP pool.
// ------------------------------------------------------------------
__device__ __forceinline__ void cas_pair(float* v, int* ix, int i, int j, int ksz) {
    int p = i ^ j;
    if (p > i) {
        bool up  = ((i & ksz) == 0);
        float a = v[i], b = v[p];
        if ((a > b) == up) {
            v[i] = b; v[p] = a;
            int t = ix[i]; ix[i] = ix[p]; ix[p] = t;
        }
    }
}

__global__ void __launch_bounds__(SORT_BLOCK)
SWD_sort_scatter_kernel(const float* __restrict__ q,
                        const float* __restrict__ k,
                        float* __restrict__ out) {
    __shared__ float qv[S_LEN];
    __shared__ float kv[S_LEN];
    __shared__ int   qi[S_LEN];
    __shared__ int   ki[S_LEN];

    const int tid = threadIdx.x;
    const int blk = blockIdx.x;          // 0 .. 2047
    const int n   = blk >> 6;            // matrix index
    const int d   = blk & (D_DIM - 1);   // dim index

    const float* qp = q + (size_t)n * S_LEN * D_DIM + d;
    const float* kp = k + (size_t)n * S_LEN * D_DIM + d;

    // ---- load columns into LDS (async, tracked by ASYNCcnt) ----
    for (int s = tid; s < S_LEN; s += SORT_BLOCK) {
#if USE_ASYNC_LDS
        __builtin_amdgcn_global_load_async_to_lds_b32(
            (gbl_i32p)(unsigned long long)(qp + (size_t)s * D_DIM),
            (lds_i32p)(unsigned int)(unsigned long long)&qv[s], 0, 0);
        __builtin_amdgcn_global_load_async_to_lds_b32(
            (gbl_i32p)(unsigned long long)(kp + (size_t)s * D_DIM),
            (lds_i32p)(unsigned int)(unsigned long long)&kv[s], 0, 0);
#else
        qv[s] = qp[(size_t)s * D_DIM];
        kv[s] = kp[(size_t)s * D_DIM];
#endif
        qi[s] = s;
        ki[s] = s;
    }
#if USE_ASYNC_LDS
    __builtin_amdgcn_s_wait_asynccnt(0);
#endif

    // ---- bitonic sort (ascending), both columns per stage ----
    // Stage with distance j: element x written by lane (x mod 512)&~j_prev,
    // read by lane (x mod 512)&~j. Same wave32 iff j_prev<32 && j<32.
    int jprev = 1;  // load handoff is wave-local (same tid residue)
    for (int ksz = 2; ksz <= S_LEN; ksz <<= 1) {
        for (int j = ksz >> 1; j > 0; j >>= 1) {
            if (j >= 32 || jprev >= 32) __syncthreads();
            else                        wave_lds_sync();
            for (int i = tid; i < S_LEN; i += SORT_BLOCK) {
                cas_pair(qv, qi, i, j, ksz);
                cas_pair(kv, ki, i, j, ksz);
            }
            jprev = j;
        }
    }
    __syncthreads();

    // ---- compute weights and scatter (atomics land in 192MB L2) ----
    float* outp = out + ((size_t)n << 20);   // n * S * S
    const float inv_d = 1.0f / (float)D_DIM;
    for (int r = tid; r < S_LEN; r += SORT_BLOCK) {
        float diff = qv[r] - kv[r];
        float w = __expf(-diff * diff) * inv_d;
        unsigned int off = ((unsigned int)qi[r] << 10) + (unsigned int)ki[r];
        atomicAdd(outp + off, w);
    }
}

// ------------------------------------------------------------------
// Kernel B: zero-fill the 134MB output with b128 stores (stays in L2,
// 192MB, so the scatter atomics in kernel A hit L2 not HBM).
// ------------------------------------------------------------------
__global__ void SWD_zero_kernel(float4* __restrict__ out4, unsigned int n4) {
    unsigned int i      = blockIdx.x * blockDim.x + threadIdx.x;
    unsigned int stride = gridDim.x * blockDim.x;
    float4 z; z.x = 0.f; z.y = 0.f; z.z = 0.f; z.w = 0.f;
    for (; i < n4; i += stride) out4[i] = z;
}

// ------------------------------------------------------------------
// Kernel C: apply mask: out[i] = mask[i] ? 0 : out[i].
// Mask bools read 16-at-a-time as uint4 (b128); all-zero words skip
// their stores. Prefetch ahead (global_prefetch_b8).
// ------------------------------------------------------------------
__global__ void SWD_mask_kernel(const uint4* __restrict__ mask128,
                                float* __restrict__ out,
                                unsigned int nvec) {
    unsigned int i      = blockIdx.x * blockDim.x + threadIdx.x;
    unsigned int stride = gridDim.x * blockDim.x;
    for (; i < nvec; i += stride) {
        __builtin_prefetch((const void*)(mask128 + i + stride), 0, 1); // speculative, OOB-safe
        uint4 m = mask128[i];
        unsigned int base = i << 4;   // 16 floats per uint4 of bools
        #pragma unroll
        for (int wsel = 0; wsel < 4; ++wsel) {
            unsigned int w = (wsel == 0) ? m.x : (wsel == 1) ? m.y : (wsel == 2) ? m.z : m.w;
            if (w) {
                unsigned int b = base + (unsigned int)(wsel << 2);
                if (w & 0x000000FFu) out[b + 0] = 0.f;
                if (w & 0x0000FF00u) out[b + 1] = 0.f;
                if (w & 0x00FF0000u) out[b + 2] = 0.f;
                if (w & 0xFF000000u) out[b + 3] = 0.f;
            }
        }
    }
}

// ------------------------------------------------------------------
extern "C" void kernel_launch(void* const* d_in, const int* in_sizes, int n_in,
                              void* d_out, int out_size, void* d_ws, size_t ws_size,
                              hipStream_t stream) {
    const float* q    = (const float*)d_in[0];
    const float* k    = (const float*)d_in[1];
    const uint4* mask = (const uint4*)d_in[2]; // 16 bools per vector
    float*       out  = (float*)d_out;

    // 1) zero-init output (stream-ordered)
    SWD_zero_kernel<<<8192, BLOCK, 0, stream>>>((float4*)out, OUT_ELEMS / 4);

    // 2) per-(n,d) column sort + scatter: 32*64 = 2048 blocks
    SWD_sort_scatter_kernel<<<N_MAT * D_DIM, SORT_BLOCK, 0, stream>>>(q, k, out);

    // 3) mask application (16 bools / thread-iteration)
    SWD_mask_kernel<<<4096, BLOCK, 0, stream>>>(mask, out, OUT_ELEMS / 16);
}